// ACE_56495999812198
// MI455X (gfx1250) — compile-verified
//
#include <hip/hip_runtime.h>
#include <hip/hip_bf16.h>

#define N_NODES  10000
#define N_EDGES  100000
#define KDIM     32
#define LDIM     4
#define NLM      16
#define N_RBF    8
#define N_ELEM   3
#define N_GRAPHS 64
#define R_CUT    5.0f
#define EPS_F    1e-9f
#define PI_F     3.14159265358979f
#define SROW     520            // node-kernel LDS row stride (floats)
#define HROW     34             // edge-kernel LDS row stride (floats, pad 32->34)
#define NODE_FEAT (NLM*KDIM)    // 512
#define WPB      4              // waves per block in edge kernels
#define EPW      16             // edges per wave

typedef __attribute__((ext_vector_type(2))) float v2f;
typedef __attribute__((ext_vector_type(8))) float v8f;

__device__ __constant__ int   LM2L_d[16]  = {0,1,1,1,2,2,2,2,2,3,3,3,3,3,3,3};
__device__ __constant__ float NORM_L_d[4] = {1.0f, 0.57735026919f, 0.44721359550f, 0.37796447301f};

__device__ __forceinline__ void sph16(float x, float y, float z, float* Y) {
  const float c3=1.73205081f, c15=3.87298335f, c5h=1.11803399f, c15h=1.93649167f,
              c70q=2.09165007f, c105=10.24695077f, c42q=1.62018517f,
              c7h=1.32287566f, c105h=5.12347538f;
  float x2=x*x, y2=y*y, z2=z*z;
  Y[0]=1.0f; Y[1]=c3*x; Y[2]=c3*y; Y[3]=c3*z;
  Y[4]=c15*x*y; Y[5]=c15*y*z; Y[6]=c5h*(3.0f*z2-1.0f); Y[7]=c15*x*z; Y[8]=c15h*(x2-y2);
  Y[9]=c70q*y*(3.0f*x2-y2); Y[10]=c105*x*y*z; Y[11]=c42q*y*(5.0f*z2-1.0f);
  Y[12]=c7h*(5.0f*z2*z-3.0f*z); Y[13]=c42q*x*(5.0f*z2-1.0f); Y[14]=c105h*z*(x2-y2);
  Y[15]=c70q*x*(x2-3.0f*y2);
}

__device__ __forceinline__ v8f wmma_f32(v2f a, v2f b, v8f c) {
  return __builtin_amdgcn_wmma_f32_16x16x4_f32(false, a, false, b, (short)0, c, false, false);
}

// ---------------------------------------------------------------------------
// Edge forward: wave-cooperative. 16 edges/wave; radial MLP on WMMA;
// coalesced 32-wide atomic scatter into A_raw[N,16,32].
// ---------------------------------------------------------------------------
__global__ void __launch_bounds__(WPB*32)
edge_fwd_kernel(const float* __restrict__ pos, const int* __restrict__ ei,
                const float* __restrict__ shifts, const int* __restrict__ species,
                const float* __restrict__ Wz,
                const float* __restrict__ rW1, const float* __restrict__ rb1,
                const float* __restrict__ rW2, const float* __restrict__ rb2,
                const float* __restrict__ rW3, const float* __restrict__ rb3,
                float* __restrict__ A_raw)
{
  __shared__ float sXR[WPB][EPW*10];     // bessel features, stride 10
  __shared__ float sY [WPB][EPW*16];     // spherical harmonics (masked)
  __shared__ float sH1[WPB][EPW*HROW];
  __shared__ float sH2[WPB][EPW*HROW];
  __shared__ float sT [WPB][EPW*HROW];   // Rl * Zs
  __shared__ int   sRecv[WPB][EPW];
  __shared__ int   sSpec[WPB][EPW];

  const int tid  = threadIdx.x;
  const int wid  = tid >> 5;
  const int lid  = tid & 31;
  const int lane = lid & 15;
  const int hi   = lid >> 4;
  const int ebase = blockIdx.x * (WPB*EPW) + wid*EPW;

  if (lid < EPW) {
    int e = ebase + lid;
    float valid = (e < N_EDGES) ? 1.0f : 0.0f;
    if (e >= N_EDGES) e = N_EDGES - 1;
    const int s  = ei[e];
    const int rv = ei[N_EDGES + e];
    float vx = pos[3*rv+0] - pos[3*s+0] + shifts[3*e+0];
    float vy = pos[3*rv+1] - pos[3*s+1] + shifts[3*e+1];
    float vz = pos[3*rv+2] - pos[3*s+2] + shifts[3*e+2];
    float r  = sqrtf(vx*vx + vy*vy + vz*vz + EPS_F);
    float invr = 1.0f / r;
    float x = vx*invr, y = vy*invr, z = vz*invr;
    float ttc = r * (1.0f/R_CUT);
    float msk = (r < R_CUT) ? 1.0f : 0.0f;
    float t2 = ttc*ttc, t3 = t2*ttc, t6 = t3*t3, t7 = t6*ttc, t8 = t7*ttc;
    float fc = (1.0f - 28.0f*t6 + 48.0f*t7 - 21.0f*t8) * msk;
    float invrp = 1.0f / (r + EPS_F);
#pragma unroll
    for (int n = 0; n < N_RBF; ++n) {
      float w = (float)(n+1) * (PI_F / R_CUT);
      sXR[wid][lid*10 + n] = 0.632455532f * sinf(w*r) * invrp * fc;
    }
    float Y[NLM];
    sph16(x, y, z, Y);
#pragma unroll
    for (int lm = 0; lm < NLM; ++lm) sY[wid][lid*16 + lm] = Y[lm] * valid;
    sRecv[wid][lid] = rv;
    sSpec[wid][lid] = species[s];
  }
  __syncthreads();

  for (int l = 0; l < LDIM; ++l) {
    // ---- layer 1: [16x8] @ [8x32] + bias, silu -> sH1
#pragma unroll
    for (int h = 0; h < 2; ++h) {
      const int o = h*16 + lane;
      v8f c = {0.f,0.f,0.f,0.f,0.f,0.f,0.f,0.f};
#pragma unroll
      for (int kc = 0; kc < 2; ++kc) {
        const int kb = kc*4 + hi*2;
        v2f a, b;
        a.x = sXR[wid][lane*10 + kb];     a.y = sXR[wid][lane*10 + kb + 1];
        b.x = rW1[(l*KDIM + o)*N_RBF + kb]; b.y = rW1[(l*KDIM + o)*N_RBF + kb + 1];
        c = wmma_f32(a, b, c);
      }
      const float bias = rb1[l*KDIM + o];
#pragma unroll
      for (int rr = 0; rr < 8; ++rr) {
        float pre = c[rr] + bias;
        float sg = 1.0f / (1.0f + expf(-pre));
        sH1[wid][(rr + 8*hi)*HROW + o] = pre * sg;
      }
    }
    __syncthreads();
    // ---- layer 2: [16x32] @ [32x32] + bias, silu -> sH2
#pragma unroll
    for (int h = 0; h < 2; ++h) {
      const int o = h*16 + lane;
      v8f c = {0.f,0.f,0.f,0.f,0.f,0.f,0.f,0.f};
#pragma unroll
      for (int kc = 0; kc < 8; ++kc) {
        const int kb = kc*4 + hi*2;
        v2f a, b;
        a.x = sH1[wid][lane*HROW + kb];      a.y = sH1[wid][lane*HROW + kb + 1];
        b.x = rW2[(l*KDIM + o)*KDIM + kb];   b.y = rW2[(l*KDIM + o)*KDIM + kb + 1];
        c = wmma_f32(a, b, c);
      }
      const float bias = rb2[l*KDIM + o];
#pragma unroll
      for (int rr = 0; rr < 8; ++rr) {
        float pre = c[rr] + bias;
        float sg = 1.0f / (1.0f + expf(-pre));
        sH2[wid][(rr + 8*hi)*HROW + o] = pre * sg;
      }
    }
    __syncthreads();
    // ---- layer 3: [16x32] @ [32x32] + bias, * Zs -> sT
#pragma unroll
    for (int h = 0; h < 2; ++h) {
      const int o = h*16 + lane;
      v8f c = {0.f,0.f,0.f,0.f,0.f,0.f,0.f,0.f};
#pragma unroll
      for (int kc = 0; kc < 8; ++kc) {
        const int kb = kc*4 + hi*2;
        v2f a, b;
        a.x = sH2[wid][lane*HROW + kb];      a.y = sH2[wid][lane*HROW + kb + 1];
        b.x = rW3[(l*KDIM + o)*KDIM + kb];   b.y = rW3[(l*KDIM + o)*KDIM + kb + 1];
        c = wmma_f32(a, b, c);
      }
      const float bias = rb3[l*KDIM + o];
#pragma unroll
      for (int rr = 0; rr < 8; ++rr) {
        const int m = rr + 8*hi;
        const int sp = sSpec[wid][m];
        float zs = Wz[o*N_ELEM + sp];
        sT[wid][m*HROW + o] = (c[rr] + bias) * zs;
      }
    }
    __syncthreads();
    // ---- coalesced scatter: 32 lanes = one contiguous k-row per (edge,lm)
    for (int lm = l*l; lm < (l+1)*(l+1); ++lm) {
      for (int j = 0; j < EPW; ++j) {
        float val = sT[wid][j*HROW + lid] * sY[wid][j*16 + lm];
        atomicAdd(&A_raw[(size_t)sRecv[wid][j]*NODE_FEAT + lm*KDIM + lid], val);
      }
    }
    __syncthreads();
  }
}

// ---------------------------------------------------------------------------
// Node kernel: WMMA f32 mix (fwd), invariants -> energy, dE/dA_mix,
// WMMA backward mix -> G written IN PLACE over A_raw. One wave / 16 nodes.
// ---------------------------------------------------------------------------
__global__ void __launch_bounds__(32)
node_kernel(float* __restrict__ AG,
            const int* __restrict__ species, const int* __restrict__ batch,
            const float* __restrict__ mixW,  const float* __restrict__ symW1,
            const float* __restrict__ symW2, const float* __restrict__ eW,
            const float* __restrict__ ebp,   const float* __restrict__ e0W,
            const float* __restrict__ e0bp,  float* __restrict__ Ebatch)
{
  __shared__ __align__(16) float sA[16*SROW];
  __shared__ __align__(16) float sM[16*SROW];

  const int tid  = threadIdx.x;
  const int lane = tid & 15;
  const int hi   = tid >> 4;
  const int nodeBase = blockIdx.x * 16;
  const float invSqrtK = 0.17677669529f;

  const float4* gsrc = (const float4*)(AG + (size_t)nodeBase * NODE_FEAT);
#pragma unroll
  for (int it = 0; it < 64; ++it) {
    int idx = tid + 32*it;
    int nd  = idx >> 7;
    int off = idx & 127;
    *(float4*)(sA + nd*SROW + off*4) = gsrc[idx];
  }
  __syncthreads();

  for (int lm = 0; lm < NLM; ++lm) {
    const int l = LM2L_d[lm];
#pragma unroll
    for (int h = 0; h < 2; ++h) {
      v8f c = {0.f,0.f,0.f,0.f,0.f,0.f,0.f,0.f};
      const int o = h*16 + lane;
#pragma unroll
      for (int kc = 0; kc < 8; ++kc) {
        const int kb = kc*4 + hi*2;
        v2f a, b;
        a.x = sA[lane*SROW + lm*KDIM + kb]; a.y = sA[lane*SROW + lm*KDIM + kb + 1];
        b.x = mixW[(l*KDIM + o)*KDIM + kb]; b.y = mixW[(l*KDIM + o)*KDIM + kb + 1];
        c = wmma_f32(a, b, c);
      }
#pragma unroll
      for (int rr = 0; rr < 8; ++rr)
        sM[(rr + 8*hi)*SROW + lm*KDIM + h*16 + lane] = c[rr] * invSqrtK;
    }
  }
  __syncthreads();

  if (tid < 16) {
    const int n    = nodeBase + tid;
    const int spec = species[n];
    float E = 0.0f;
    for (int k = 0; k < KDIM; ++k) {
      float a0 = sM[tid*SROW + k];
      float bk = symW1[spec*KDIM + k] * a0;
#pragma unroll
      for (int l = 0; l < LDIM; ++l) {
        float pq = 0.0f;
        for (int lm = l*l; lm < (l+1)*(l+1); ++lm) {
          float v = sM[tid*SROW + lm*KDIM + k];
          pq += v*v;
        }
        bk += pq * NORM_L_d[l] * symW2[(spec*LDIM + l)*KDIM + k];
      }
      E += bk * eW[k];
    }
    E += ebp[0] + e0W[spec] + e0bp[0];
    atomicAdd(&Ebatch[batch[n]], E);

    for (int lm = 0; lm < NLM; ++lm) {
      const int l = LM2L_d[lm];
      const float nl2w = 2.0f * NORM_L_d[l];
      for (int k = 0; k < KDIM; ++k) {
        float v = sM[tid*SROW + lm*KDIM + k];
        float g = eW[k] * (((lm == 0) ? symW1[spec*KDIM + k] : 0.0f)
                           + nl2w * symW2[(spec*LDIM + l)*KDIM + k] * v);
        sM[tid*SROW + lm*KDIM + k] = g;
      }
    }
  }
  __syncthreads();

  for (int lm = 0; lm < NLM; ++lm) {
    const int l = LM2L_d[lm];
#pragma unroll
    for (int h = 0; h < 2; ++h) {
      v8f c = {0.f,0.f,0.f,0.f,0.f,0.f,0.f,0.f};
      const int ncol = h*16 + lane;
#pragma unroll
      for (int kc = 0; kc < 8; ++kc) {
        const int kb = kc*4 + hi*2;
        v2f a, b;
        a.x = sM[lane*SROW + lm*KDIM + kb]; a.y = sM[lane*SROW + lm*KDIM + kb + 1];
        b.x = mixW[(l*KDIM + kb    )*KDIM + ncol];
        b.y = mixW[(l*KDIM + kb + 1)*KDIM + ncol];
        c = wmma_f32(a, b, c);
      }
#pragma unroll
      for (int rr = 0; rr < 8; ++rr)
        AG[(size_t)(nodeBase + rr + 8*hi)*NODE_FEAT + lm*KDIM + ncol] = c[rr] * invSqrtK;
    }
  }
}

// ---------------------------------------------------------------------------
// Edge backward: wave-cooperative; forward recompute + backward MLP on WMMA;
// coalesced G gather; chain rule -> forces.
// ---------------------------------------------------------------------------
__global__ void __launch_bounds__(WPB*32)
edge_bwd_kernel(const float* __restrict__ pos, const int* __restrict__ ei,
                const float* __restrict__ shifts, const int* __restrict__ species,
                const float* __restrict__ Wz,
                const float* __restrict__ rW1, const float* __restrict__ rb1,
                const float* __restrict__ rW2, const float* __restrict__ rb2,
                const float* __restrict__ rW3, const float* __restrict__ rb3,
                const float* __restrict__ G, float* __restrict__ F)
{
  __shared__ float sXR[WPB][EPW*10];
  __shared__ float sY [WPB][EPW*16];
  __shared__ float sDY[WPB][EPW*16];
  __shared__ float sS1[WPB][EPW*HROW];
  __shared__ float sH1[WPB][EPW*HROW];
  __shared__ float sS2[WPB][EPW*HROW];
  __shared__ float sH2[WPB][EPW*HROW];
  __shared__ float sT [WPB][EPW*HROW];   // fwd: Rl*Zs ; later: dh1
  __shared__ float sDR[WPB][EPW*HROW];   // dE/dRl
  __shared__ float sGw[WPB][EPW*HROW];   // gathered G row ; later: dh2
  __shared__ int   sRecv[WPB][EPW];
  __shared__ int   sSpec[WPB][EPW];

  const int tid  = threadIdx.x;
  const int wid  = tid >> 5;
  const int lid  = tid & 31;
  const int lane = lid & 15;
  const int hi   = lid >> 4;
  const int ebase = blockIdx.x * (WPB*EPW) + wid*EPW;

  float validf = 0.0f, x = 0.f, y = 0.f, z = 0.f, invr = 1.f;
  int   sidx = 0, ridx = 0;
  float dxr_dr[N_RBF];
  float dxr[N_RBF];
#pragma unroll
  for (int n = 0; n < N_RBF; ++n) { dxr_dr[n] = 0.f; dxr[n] = 0.f; }

  if (lid < EPW) {
    int e = ebase + lid;
    validf = (e < N_EDGES) ? 1.0f : 0.0f;
    if (e >= N_EDGES) e = N_EDGES - 1;
    sidx = ei[e];
    ridx = ei[N_EDGES + e];
    float vx = pos[3*ridx+0] - pos[3*sidx+0] + shifts[3*e+0];
    float vy = pos[3*ridx+1] - pos[3*sidx+1] + shifts[3*e+1];
    float vz = pos[3*ridx+2] - pos[3*sidx+2] + shifts[3*e+2];
    float r  = sqrtf(vx*vx + vy*vy + vz*vz + EPS_F);
    invr = 1.0f / r;
    x = vx*invr; y = vy*invr; z = vz*invr;
    float ttc = r * (1.0f/R_CUT);
    float msk = (r < R_CUT) ? 1.0f : 0.0f;
    float t2 = ttc*ttc, t3 = t2*ttc, t5 = t2*t3, t6 = t3*t3, t7 = t6*ttc, t8 = t7*ttc;
    float fc  = (1.0f - 28.0f*t6 + 48.0f*t7 - 21.0f*t8) * msk;
    float fcp = (-168.0f*t5 + 336.0f*t6 - 168.0f*t7) * (1.0f/R_CUT) * msk;
    float rp = r + EPS_F;
    float invrp = 1.0f / rp;
#pragma unroll
    for (int n = 0; n < N_RBF; ++n) {
      float w  = (float)(n+1) * (PI_F / R_CUT);
      float sn = sinf(w*r), cn = cosf(w*r);
      sXR[wid][lid*10 + n] = 0.632455532f * sn * invrp * fc;
      dxr_dr[n] = 0.632455532f * ((w*cn*rp - sn)*invrp*invrp*fc + sn*invrp*fcp);
    }
    float Y[NLM];
    sph16(x, y, z, Y);
#pragma unroll
    for (int lm = 0; lm < NLM; ++lm) sY[wid][lid*16 + lm] = Y[lm] * validf;
    sRecv[wid][lid] = ridx;
    sSpec[wid][lid] = species[sidx];
  }
  __syncthreads();

  for (int l = 0; l < LDIM; ++l) {
    // ---- forward layer 1 (save sigmoid) ----
#pragma unroll
    for (int h = 0; h < 2; ++h) {
      const int o = h*16 + lane;
      v8f c = {0.f,0.f,0.f,0.f,0.f,0.f,0.f,0.f};
#pragma unroll
      for (int kc = 0; kc < 2; ++kc) {
        const int kb = kc*4 + hi*2;
        v2f a, b;
        a.x = sXR[wid][lane*10 + kb];       a.y = sXR[wid][lane*10 + kb + 1];
        b.x = rW1[(l*KDIM + o)*N_RBF + kb]; b.y = rW1[(l*KDIM + o)*N_RBF + kb + 1];
        c = wmma_f32(a, b, c);
      }
      const float bias = rb1[l*KDIM + o];
#pragma unroll
      for (int rr = 0; rr < 8; ++rr) {
        const int m = rr + 8*hi;
        float pre = c[rr] + bias;
        float sg = 1.0f / (1.0f + expf(-pre));
        sS1[wid][m*HROW + o] = sg;
        sH1[wid][m*HROW + o] = pre * sg;
      }
    }
    __syncthreads();
    // ---- forward layer 2 (save sigmoid) ----
#pragma unroll
    for (int h = 0; h < 2; ++h) {
      const int o = h*16 + lane;
      v8f c = {0.f,0.f,0.f,0.f,0.f,0.f,0.f,0.f};
#pragma unroll
      for (int kc = 0; kc < 8; ++kc) {
        const int kb = kc*4 + hi*2;
        v2f a, b;
        a.x = sH1[wid][lane*HROW + kb];    a.y = sH1[wid][lane*HROW + kb + 1];
        b.x = rW2[(l*KDIM + o)*KDIM + kb]; b.y = rW2[(l*KDIM + o)*KDIM + kb + 1];
        c = wmma_f32(a, b, c);
      }
      const float bias = rb2[l*KDIM + o];
#pragma unroll
      for (int rr = 0; rr < 8; ++rr) {
        const int m = rr + 8*hi;
        float pre = c[rr] + bias;
        float sg = 1.0f / (1.0f + expf(-pre));
        sS2[wid][m*HROW + o] = sg;
        sH2[wid][m*HROW + o] = pre * sg;
      }
    }
    __syncthreads();
    // ---- forward layer 3 -> t = Rl*Zs ----
#pragma unroll
    for (int h = 0; h < 2; ++h) {
      const int o = h*16 + lane;
      v8f c = {0.f,0.f,0.f,0.f,0.f,0.f,0.f,0.f};
#pragma unroll
      for (int kc = 0; kc < 8; ++kc) {
        const int kb = kc*4 + hi*2;
        v2f a, b;
        a.x = sH2[wid][lane*HROW + kb];    a.y = sH2[wid][lane*HROW + kb + 1];
        b.x = rW3[(l*KDIM + o)*KDIM + kb]; b.y = rW3[(l*KDIM + o)*KDIM + kb + 1];
        c = wmma_f32(a, b, c);
      }
      const float bias = rb3[l*KDIM + o];
#pragma unroll
      for (int rr = 0; rr < 8; ++rr) {
        const int m = rr + 8*hi;
        const int sp = sSpec[wid][m];
        sT[wid][m*HROW + o] = (c[rr] + bias) * Wz[o*N_ELEM + sp];
      }
    }
    __syncthreads();
    // ---- gather G, accumulate dR (regs) and dY (LDS bounce) ----
    float dRacc[EPW];
#pragma unroll
    for (int j = 0; j < EPW; ++j) dRacc[j] = 0.0f;
    for (int lm = l*l; lm < (l+1)*(l+1); ++lm) {
#pragma unroll
      for (int j = 0; j < EPW; ++j) {
        const int rvj = sRecv[wid][j];
        float g  = G[(size_t)rvj*NODE_FEAT + lm*KDIM + lid];
        float yv = sY[wid][j*16 + lm];
        const int sp = sSpec[wid][j];
        float zs = Wz[lid*N_ELEM + sp];
        dRacc[j] += g * yv * zs;
        sGw[wid][j*HROW + lid] = g;
      }
      __syncthreads();
      if (lid < EPW) {
        float acc = 0.0f;
        for (int k = 0; k < KDIM; ++k)
          acc += sGw[wid][lid*HROW + k] * sT[wid][lid*HROW + k];
        sDY[wid][lid*16 + lm] = acc;
      }
      __syncthreads();
    }
#pragma unroll
    for (int j = 0; j < EPW; ++j) sDR[wid][j*HROW + lid] = dRacc[j];
    __syncthreads();
    // ---- dh2 = (dR @ rW3) * silu'(pre2) -> sGw ----
#pragma unroll
    for (int h = 0; h < 2; ++h) {
      const int o = h*16 + lane;
      v8f c = {0.f,0.f,0.f,0.f,0.f,0.f,0.f,0.f};
#pragma unroll
      for (int kc = 0; kc < 8; ++kc) {
        const int kb = kc*4 + hi*2;
        v2f a, b;
        a.x = sDR[wid][lane*HROW + kb];  a.y = sDR[wid][lane*HROW + kb + 1];
        b.x = rW3[(l*KDIM + kb    )*KDIM + o];
        b.y = rW3[(l*KDIM + kb + 1)*KDIM + o];
        c = wmma_f32(a, b, c);
      }
#pragma unroll
      for (int rr = 0; rr < 8; ++rr) {
        const int m = rr + 8*hi;
        float s2 = sS2[wid][m*HROW + o];
        float h2 = sH2[wid][m*HROW + o];
        sGw[wid][m*HROW + o] = c[rr] * (s2 + h2*(1.0f - s2));
      }
    }
    __syncthreads();
    // ---- dh1 = (dh2 @ rW2) * silu'(pre1) -> sT ----
#pragma unroll
    for (int h = 0; h < 2; ++h) {
      const int o = h*16 + lane;
      v8f c = {0.f,0.f,0.f,0.f,0.f,0.f,0.f,0.f};
#pragma unroll
      for (int kc = 0; kc < 8; ++kc) {
        const int kb = kc*4 + hi*2;
        v2f a, b;
        a.x = sGw[wid][lane*HROW + kb];  a.y = sGw[wid][lane*HROW + kb + 1];
        b.x = rW2[(l*KDIM + kb    )*KDIM + o];
        b.y = rW2[(l*KDIM + kb + 1)*KDIM + o];
        c = wmma_f32(a, b, c);
      }
#pragma unroll
      for (int rr = 0; rr < 8; ++rr) {
        const int m = rr + 8*hi;
        float s1 = sS1[wid][m*HROW + o];
        float h1 = sH1[wid][m*HROW + o];
        sT[wid][m*HROW + o] = c[rr] * (s1 + h1*(1.0f - s1));
      }
    }
    __syncthreads();
    // ---- dxr += dh1 @ rW1^T (per-edge lanes, small 32x8) ----
    if (lid < EPW) {
#pragma unroll
      for (int q = 0; q < N_RBF; ++q) {
        float acc = 0.0f;
        for (int k = 0; k < KDIM; ++k)
          acc += sT[wid][lid*HROW + k] * rW1[(l*KDIM + k)*N_RBF + q];
        dxr[q] += acc;
      }
    }
    __syncthreads();
  }

  // ---- finalize: chain rule to positions (per-edge lanes) ----
  if (lid < EPW) {
    float dr = 0.0f;
#pragma unroll
    for (int n = 0; n < N_RBF; ++n) dr += dxr[n] * dxr_dr[n];

    const float* dY = &sDY[wid][lid*16];
    const float c3=1.73205081f, c15=3.87298335f, c5h=1.11803399f, c15h=1.93649167f,
                c70q=2.09165007f, c105=10.24695077f, c42q=1.62018517f,
                c7h=1.32287566f, c105h=5.12347538f;
    float x2=x*x, y2=y*y, z2=z*z;
    float dux = c3*dY[1] + c15*y*dY[4] + c15*z*dY[7] + 2.0f*c15h*x*dY[8]
              + c70q*6.0f*x*y*dY[9] + c105*y*z*dY[10] + c42q*(5.0f*z2-1.0f)*dY[13]
              + c105h*2.0f*x*z*dY[14] + c70q*(3.0f*x2-3.0f*y2)*dY[15];
    float duy = c3*dY[2] + c15*x*dY[4] + c15*z*dY[5] - 2.0f*c15h*y*dY[8]
              + c70q*(3.0f*x2-3.0f*y2)*dY[9] + c105*x*z*dY[10]
              + c42q*(5.0f*z2-1.0f)*dY[11] - c105h*2.0f*y*z*dY[14]
              - c70q*6.0f*x*y*dY[15];
    float duz = c3*dY[3] + c15*y*dY[5] + 6.0f*c5h*z*dY[6] + c15*x*dY[7]
              + c105*x*y*dY[10] + c42q*10.0f*y*z*dY[11]
              + c7h*(15.0f*z2-3.0f)*dY[12] + c42q*10.0f*x*z*dY[13]
              + c105h*(x2-y2)*dY[14];

    float dot = x*dux + y*duy + z*duz;
    float gx = (dr*x + (dux - dot*x)*invr) * validf;
    float gy = (dr*y + (duy - dot*y)*invr) * validf;
    float gz = (dr*z + (duz - dot*z)*invr) * validf;

    atomicAdd(&F[3*sidx+0],  gx);
    atomicAdd(&F[3*sidx+1],  gy);
    atomicAdd(&F[3*sidx+2],  gz);
    atomicAdd(&F[3*ridx+0], -gx);
    atomicAdd(&F[3*ridx+1], -gy);
    atomicAdd(&F[3*ridx+2], -gz);
  }
}

// ---------------------------------------------------------------------------
extern "C" void kernel_launch(void* const* d_in, const int* in_sizes, int n_in,
                              void* d_out, int out_size, void* d_ws, size_t ws_size,
                              hipStream_t stream)
{
  const float* pos     = (const float*)d_in[0];
  const int*   ei      = (const int*)  d_in[1];
  const float* shifts  = (const float*)d_in[2];
  const int*   species = (const int*)  d_in[3];
  const int*   batch   = (const int*)  d_in[4];
  const float* Wz      = (const float*)d_in[5];
  const float* rW1     = (const float*)d_in[6];
  const float* rb1     = (const float*)d_in[7];
  const float* rW2     = (const float*)d_in[8];
  const float* rb2     = (const float*)d_in[9];
  const float* rW3     = (const float*)d_in[10];
  const float* rb3     = (const float*)d_in[11];
  const float* mixW    = (const float*)d_in[12];
  const float* symW1   = (const float*)d_in[13];
  const float* symW2   = (const float*)d_in[14];
  const float* eW      = (const float*)d_in[15];
  const float* ebp     = (const float*)d_in[16];
  const float* e0W     = (const float*)d_in[17];
  const float* e0bp    = (const float*)d_in[18];

  float* out = (float*)d_out;
  float* Eb  = out;
  float* F   = out + N_GRAPHS;
  float* AG  = (float*)d_ws;   // A_raw forward, overwritten with G by node_kernel

  hipMemsetAsync(AG, 0, (size_t)N_NODES * NODE_FEAT * sizeof(float), stream);
  hipMemsetAsync(out, 0, (size_t)out_size * sizeof(float), stream);

  const int edgeBlocks = (N_EDGES + WPB*EPW - 1) / (WPB*EPW);

  edge_fwd_kernel<<<edgeBlocks, WPB*32, 0, stream>>>(
      pos, ei, shifts, species, Wz, rW1, rb1, rW2, rb2, rW3, rb3, AG);

  node_kernel<<<N_NODES/16, 32, 0, stream>>>(
      AG, species, batch, mixW, symW1, symW2, eW, ebp, e0W, e0bp, Eb);

  edge_bwd_kernel<<<edgeBlocks, WPB*32, 0, stream>>>(
      pos, ei, shifts, species, Wz, rW1, rb1, rW2, rb2, rW3, rb3, AG, F);
}